// healthCRL_85349590106293
// MI455X (gfx1250) — compile-verified
//
#include <hip/hip_runtime.h>

typedef float v2f __attribute__((ext_vector_type(2)));
typedef float v8f __attribute__((ext_vector_type(8)));

#define N_NODES 50000
#define N_EDGES 1600000
#define DIM     128
#define NLAYERS 3
#define NGRAPHS 512
#define EDIM    (NLAYERS * DIM)   // 384
#define BN_EPS  1e-5f

#if __has_builtin(__builtin_amdgcn_global_load_async_to_lds_b32)
#define HAVE_ASYNC_LDS 1
#else
#define HAVE_ASYNC_LDS 0
#endif

typedef __attribute__((address_space(1))) int as1_int;
typedef __attribute__((address_space(3))) int as3_int;

__device__ __forceinline__ void wait_async_zero() {
#if __has_builtin(__builtin_amdgcn_s_wait_asynccnt)
    __builtin_amdgcn_s_wait_asynccnt(0);
#else
    asm volatile("s_wait_asynccnt 0" ::: "memory");
#endif
}

// ---------------------------------------------------------------------------
// Utility kernels
// ---------------------------------------------------------------------------
__global__ void zero_f32(float* __restrict__ p, int n) {
    int i = blockIdx.x * blockDim.x + threadIdx.x;
    if (i < n) p[i] = 0.0f;
}

// dst[r*cols + c] = src[r*lds + c]   (init agg = h)
__global__ void copy_strided(const float* __restrict__ src, int lds,
                             float* __restrict__ dst, int rows, int cols) {
    int i = blockIdx.x * blockDim.x + threadIdx.x;
    if (i >= rows * cols) return;
    int r = i / cols, c = i - r * cols;
    dst[(size_t)r * cols + c] = src[(size_t)r * lds + c];
}

// agg[dst[e]] += h[src[e]]  — 32 threads (one wave) per edge, float4 per lane.
__global__ void scatter_edges(const float* __restrict__ h, int ldh,
                              const int* __restrict__ srcI,
                              const int* __restrict__ dstI,
                              float* __restrict__ agg) {
    int idx = blockIdx.x * blockDim.x + threadIdx.x;
    int e = idx >> 5;
    if (e >= N_EDGES) return;
    int q = (idx & 31) << 2;                 // feature offset 0..124
    int s = srcI[e], d = dstI[e];
    const float4 v = *reinterpret_cast<const float4*>(h + (size_t)s * ldh + q);
    float* out = agg + (size_t)d * DIM + q;
    atomicAdd(out + 0, v.x);
    atomicAdd(out + 1, v.y);
    atomicAdd(out + 2, v.z);
    atomicAdd(out + 3, v.w);
}

// ---------------------------------------------------------------------------
// WMMA GEMM: C[M,Nc] = relu?(A[M,Kc] @ W[Kc,Nc] + bias)
// Compile-time Kc/Nc (strides). Block = 256 threads = 8 waves.
// Block tile: 128 rows x 64 cols. Wave tile: 16 rows x 64 cols (4 WMMA tiles).
// W is staged per 128-deep K-chunk into LDS, transposed (column-major per
// output column) with PAD=K_T+4 so ds_load_b64 fragments are bank-conflict-free.
// Tail row-tiles handled by inactive waves that still participate in
// staging/barriers (EXEC all-ones through WMMA).
// ---------------------------------------------------------------------------
template <int Kc, int Nc>
__global__ __launch_bounds__(256)
void wmma_gemm_bias(const float* __restrict__ A,
                    const float* __restrict__ W,
                    const float* __restrict__ bias,
                    float* __restrict__ C,
                    int M, int do_relu) {
    constexpr int KT   = 128;             // K-chunk staged per round (Kc % KT == 0)
    constexpr int PAD  = KT + 4;          // word stride between staged columns
    constexpr int COLB = 64;              // block column-tile width
    constexpr int colBlocks = Nc / COLB;

    __shared__ float ldsB[COLB * PAD];    // 33,792 B

    const int cb  = blockIdx.x % colBlocks;
    const int rb  = blockIdx.x / colBlocks;
    const int tn0 = cb * COLB;

    const int waveId  = threadIdx.x >> 5;
    const int lane    = threadIdx.x & 31;
    const int half    = lane >> 4;        // 0 or 1
    const int l       = lane & 15;
    const int rowTile = rb * 8 + waveId;
    const int tilesM  = M >> 4;
    const bool active = rowTile < tilesM;
    const int tm      = active ? (rowTile << 4) : 0;

    const float* Arow = A + (size_t)(tm + l) * Kc;  // this lane's A row
    const float* B0   = &ldsB[l * PAD];

    v8f acc[4] = {};

    for (int kk = 0; kk < Kc; kk += KT) {
        __syncthreads();                  // protect LDS from previous chunk's readers

        // ---- stage W[kk..kk+KT-1, tn0..tn0+63] transposed into LDS ----
        for (int e = threadIdx.x; e < COLB * KT; e += 256) {
            const int k = e >> 6;         // 0..KT-1
            const int n = e & 63;         // 0..63  (coalesced along n)
#if HAVE_ASYNC_LDS
            __builtin_amdgcn_global_load_async_to_lds_b32(
                (as1_int*)(W + (size_t)(kk + k) * Nc + tn0 + n),
                (as3_int*)&ldsB[n * PAD + k],
                0, 0);
#else
            ldsB[n * PAD + k] = W[(size_t)(kk + k) * Nc + tn0 + n];
#endif
        }
#if HAVE_ASYNC_LDS
        wait_async_zero();
#endif
        __syncthreads();

#if __has_builtin(__builtin_amdgcn_wmma_f32_16x16x4_f32)
        for (int k0 = 0; k0 < KT; k0 += 4) {
            const int ka = k0 + 2 * half;
            // A fragment (16x4): lane<16 -> K=ka,ka+1 of row tm+l; lane>=16 -> +2,+3
            const v2f a = *(const v2f*)(Arow + kk + ka);
#pragma unroll
            for (int j = 0; j < 4; ++j) {
                // B fragment (4x16): lane holds column tn0+16j+l, rows ka, ka+1
                const v2f b = *(const v2f*)(B0 + j * 16 * PAD + ka);
                acc[j] = __builtin_amdgcn_wmma_f32_16x16x4_f32(
                    /*neg_a=*/false, a, /*neg_b=*/false, b,
                    /*c_mod=*/(short)0, acc[j],
                    /*reuse_a=*/false, /*reuse_b=*/false);
            }
        }
#else
        // Scalar fallback matching the WMMA D layout
#pragma unroll
        for (int j = 0; j < 4; ++j) {
            const float* bc = &ldsB[(16 * j + l) * PAD];
#pragma unroll
            for (int r = 0; r < 8; ++r) {
                const float* ar = A + (size_t)(tm + r + 8 * half) * Kc + kk;
                float s = 0.0f;
                for (int k = 0; k < KT; ++k) s += ar[k] * bc[k];
                acc[j][r] += s;
            }
        }
#endif
    }

    if (active) {
        // D layout: VGPR r -> row tm + r + 8*half, col tn0 + 16j + l
#pragma unroll
        for (int j = 0; j < 4; ++j) {
            const int col = tn0 + 16 * j + l;
            const float bv = bias[col];
#pragma unroll
            for (int r = 0; r < 8; ++r) {
                float v = acc[j][r] + bv;
                if (do_relu) v = fmaxf(v, 0.0f);
                C[(size_t)(tm + r + 8 * half) * Nc + col] = v;
            }
        }
    }
}

// ---------------------------------------------------------------------------
// Column sums / sums-of-squares over u[M, DIM] (for BatchNorm stats)
// ---------------------------------------------------------------------------
__global__ void col_stats(const float* __restrict__ u,
                          float* __restrict__ sums, float* __restrict__ sumsq,
                          int M) {
    int c   = threadIdx.x;                   // 0..127
    int rpb = (M + gridDim.x - 1) / gridDim.x;
    int r0  = blockIdx.x * rpb;
    int r1  = min(M, r0 + rpb);
    float s = 0.0f, sq = 0.0f;
    for (int r = r0; r < r1; ++r) {
        float v = u[(size_t)r * DIM + c];
        s += v; sq += v * v;
    }
    atomicAdd(&sums[c], s);
    atomicAdd(&sumsq[c], sq);
}

// BN (training mode, biased var): out[r*ldo + c] = (u-mu)*rsqrt(var+eps)*g + b
__global__ void bn_apply(const float* __restrict__ u,
                         const float* __restrict__ sums, const float* __restrict__ sumsq,
                         const float* __restrict__ gamma, const float* __restrict__ beta,
                         float* __restrict__ out, int ldo) {
    int i = blockIdx.x * blockDim.x + threadIdx.x;
    if (i >= N_NODES * DIM) return;
    int r = i / DIM, c = i - r * DIM;
    const float invN = 1.0f / (float)N_NODES;
    float mu  = sums[c] * invN;
    float var = sumsq[c] * invN - mu * mu;
    float inv = rsqrtf(var + BN_EPS);
    out[(size_t)r * ldo + c] = (u[i] - mu) * inv * gamma[c] + beta[c];
}

// pooled[batch[r], c] += node[r, c]   (global_add_pool via L2 atomics)
__global__ void pool_sum(const float* __restrict__ node,
                         const int* __restrict__ batch,
                         float* __restrict__ pooled) {
    int i = blockIdx.x * blockDim.x + threadIdx.x;
    if (i >= N_NODES * EDIM) return;
    int r = i / EDIM, c = i - r * EDIM;
    atomicAdd(&pooled[(size_t)batch[r] * EDIM + c], node[i]);
}

// ---------------------------------------------------------------------------
// Host-side orchestration
// ---------------------------------------------------------------------------
extern "C" void kernel_launch(void* const* d_in, const int* in_sizes, int n_in,
                              void* d_out, int out_size, void* d_ws, size_t ws_size,
                              hipStream_t stream) {
    const float* x     = (const float*)d_in[0];
    const int*   edge  = (const int*)  d_in[1];   // [2, E] row-major
    const int*   batch = (const int*)  d_in[2];
    const float* W1    = (const float*)d_in[3];   // [L,128,128]
    const float* b1    = (const float*)d_in[4];   // [L,128]
    const float* W2    = (const float*)d_in[5];
    const float* b2    = (const float*)d_in[6];
    const float* gamma = (const float*)d_in[7];
    const float* beta  = (const float*)d_in[8];
    const float* Wp1   = (const float*)d_in[9];   // [384,384]
    const float* bp1   = (const float*)d_in[10];
    const float* Wp2   = (const float*)d_in[11];
    const float* bp2   = (const float*)d_in[12];

    const int* srcI = edge;             // edge_index[0]
    const int* dstI = edge + N_EDGES;   // edge_index[1]

    // Workspace layout (floats)
    float* ws     = (float*)d_ws;
    float* agg    = ws;                                   // [N,128] (also reused as u)
    float* t      = agg    + (size_t)N_NODES * DIM;       // [N,128]
    float* node   = t      + (size_t)N_NODES * DIM;       // [N,384] concat of layer outs
    float* pooled = node   + (size_t)N_NODES * EDIM;      // [G,384]
    float* hidden = pooled + (size_t)NGRAPHS * EDIM;      // [G,384]
    float* stats  = hidden + (size_t)NGRAPHS * EDIM;      // 256 (sums | sumsq)

    const int TPB    = 256;
    const int nElem  = N_NODES * DIM;                     // 6.4M
    const int gElem  = (nElem + TPB - 1) / TPB;
    const int gScat  = (N_EDGES * 32 + TPB - 1) / TPB;

    // GEMM grids: gridX = rowBlocks * colBlocks (colBlocks = Nc/64)
    const int rowBlkL = ((N_NODES / 16) + 7) / 8;         // 391
    const int gridL   = rowBlkL * (DIM / 64);             // 782
    const int rowBlkP = ((NGRAPHS / 16) + 7) / 8;         // 4
    const int gridP   = rowBlkP * (EDIM / 64);            // 24

    for (int i = 0; i < NLAYERS; ++i) {
        const float* h   = (i == 0) ? x   : node + (size_t)(i - 1) * DIM;
        int          ldh = (i == 0) ? DIM : EDIM;

        // agg = h  (GIN eps=0 self term)
        copy_strided<<<gElem, TPB, 0, stream>>>(h, ldh, agg, N_NODES, DIM);
        // agg[dst] += h[src]
        scatter_edges<<<gScat, TPB, 0, stream>>>(h, ldh, srcI, dstI, agg);
        // t = relu(agg @ W1[i] + b1[i])
        wmma_gemm_bias<DIM, DIM><<<gridL, TPB, 0, stream>>>(
            agg, W1 + (size_t)i * DIM * DIM, b1 + (size_t)i * DIM, t, N_NODES, 1);
        // u = relu(t @ W2[i] + b2[i])   (stored back into agg)
        wmma_gemm_bias<DIM, DIM><<<gridL, TPB, 0, stream>>>(
            t, W2 + (size_t)i * DIM * DIM, b2 + (size_t)i * DIM, agg, N_NODES, 1);
        // BN stats + apply -> node[:, i*128 : (i+1)*128]
        zero_f32<<<1, 256, 0, stream>>>(stats, 256);
        col_stats<<<256, DIM, 0, stream>>>(agg, stats, stats + DIM, N_NODES);
        bn_apply<<<gElem, TPB, 0, stream>>>(agg, stats, stats + DIM,
                                            gamma + (size_t)i * DIM, beta + (size_t)i * DIM,
                                            node + (size_t)i * DIM, EDIM);
    }

    // global_add_pool
    const int pElem = NGRAPHS * EDIM;                     // 196608
    zero_f32<<<(pElem + TPB - 1) / TPB, TPB, 0, stream>>>(pooled, pElem);
    pool_sum<<<(N_NODES * EDIM + TPB - 1) / TPB, TPB, 0, stream>>>(node, batch, pooled);

    // proj head: z = relu(pooled @ Wp1 + bp1) @ Wp2 + bp2
    wmma_gemm_bias<EDIM, EDIM><<<gridP, TPB, 0, stream>>>(pooled, Wp1, bp1,
                                                          hidden, NGRAPHS, 1);
    wmma_gemm_bias<EDIM, EDIM><<<gridP, TPB, 0, stream>>>(hidden, Wp2, bp2,
                                                          (float*)d_out, NGRAPHS, 0);
}